// DM_gcn_85667417686487
// MI455X (gfx1250) — compile-verified
//
#include <hip/hip_runtime.h>
#include <hip/hip_bf16.h>

// Problem constants (from the reference)
#define DIAGNUM 50000
#define MEDNUM  20000
#define PRONUM  40000
#define FEATDIM 128
#define N1      (DIAGNUM + MEDNUM)   // 70000
#define N2      (PRONUM  + MEDNUM)   // 60000
#define NNZ1    1120000
#define NNZ2    960000

#define TILE            256          // edges staged per block iteration
#define EDGES_PER_WAVE  32           // TILE / 8 waves

// ---- CDNA5 feature guards (compile-only environment; keep fallbacks) ----
#if defined(__has_builtin)
#  if __has_builtin(__builtin_amdgcn_global_load_async_to_lds_b32) && \
      __has_builtin(__builtin_amdgcn_s_wait_asynccnt)
#    define USE_ASYNC_LDS 1
#  else
#    define USE_ASYNC_LDS 0
#  endif
#  if __has_builtin(__builtin_amdgcn_global_atomic_fadd_f32)
#    define USE_FADD_BUILTIN 1
#  else
#    define USE_FADD_BUILTIN 0
#  endif
#else
#  define USE_ASYNC_LDS 0
#  define USE_FADD_BUILTIN 0
#endif

// cpol: TH[2:0] in the cache-policy immediate; 1 = non-temporal (streaming
// edge data must not evict the L2-resident embedding tables / scatter target)
#define CPOL_NT 1

// Builtin pointer parameter types (from hipcc diagnostics):
//   global_load_async_to_lds_b32(int AS1*, int AS3*, imm, imm)
typedef __attribute__((address_space(1))) int   gint_t;
typedef __attribute__((address_space(3))) int   lint_t;
typedef __attribute__((address_space(1))) float gfloat_t;

__device__ __forceinline__ void atomAddF(float* p, float v) {
#if USE_FADD_BUILTIN
  __builtin_amdgcn_global_atomic_fadd_f32((gfloat_t*)(void*)p, v);
#else
  atomicAdd(p, v);
#endif
}

// -------------------------------------------------------------------------
// COO scatter SpMM: S[row,:] += val * x[col,:]
// x[col] = (col < nA) ? srcA[col] : srcB[col - nA]   (concat of two tables)
// One wave per edge: 32 lanes x float4 covers the 128-wide feature row.
// Edge triplets are staged per-tile into LDS with CDNA5 async loads (NT).
__global__ __launch_bounds__(256)
void spmm_scatter(const int*   __restrict__ rows,
                  const int*   __restrict__ cols,
                  const float* __restrict__ vals,
                  int nnz,
                  const float* __restrict__ srcA, int nA,
                  const float* __restrict__ srcB,
                  float* __restrict__ S)
{
  __shared__ int   s_rows[TILE];
  __shared__ int   s_cols[TILE];
  __shared__ float s_vals[TILE];

  const int tid  = threadIdx.x;
  const int lane = tid & 31;
  const int wave = tid >> 5;
  const int c4   = lane * 4;   // this lane's feature offset

  for (long tile0 = (long)blockIdx.x * TILE; tile0 < nnz;
       tile0 += (long)gridDim.x * TILE) {
    int nEdges = nnz - (int)tile0;
    if (nEdges > TILE) nEdges = TILE;

    // ---- stage this tile's edge data into LDS (CDNA5 async copy, NT) ----
    if (tid < nEdges) {
      int g = (int)tile0 + tid;
#if USE_ASYNC_LDS
      __builtin_amdgcn_global_load_async_to_lds_b32(
          (gint_t*)(rows + g), (lint_t*)&s_rows[tid], 0, CPOL_NT);
      __builtin_amdgcn_global_load_async_to_lds_b32(
          (gint_t*)(cols + g), (lint_t*)&s_cols[tid], 0, CPOL_NT);
      __builtin_amdgcn_global_load_async_to_lds_b32(
          (gint_t*)(void*)(vals + g), (lint_t*)(void*)&s_vals[tid], 0, CPOL_NT);
#else
      s_rows[tid] = rows[g];
      s_cols[tid] = cols[g];
      s_vals[tid] = vals[g];
#endif
    }
#if USE_ASYNC_LDS
    __builtin_amdgcn_s_wait_asynccnt(0);
#endif
    __syncthreads();

    // ---- each wave processes its 32 edges of the tile ----
    int eBeg = wave * EDGES_PER_WAVE;
    int eEnd = eBeg + EDGES_PER_WAVE;
    if (eEnd > nEdges) eEnd = nEdges;

    for (int e = eBeg; e < eEnd; ++e) {
      int   row = s_rows[e];
      int   col = s_cols[e];
      float v   = s_vals[e];

      // prefetch next edge's embedding row (random gather -> hide latency)
      if (e + 1 < eEnd) {
        int nc = s_cols[e + 1];
        const float* np = (nc < nA) ? (srcA + (long)nc * FEATDIM)
                                    : (srcB + (long)(nc - nA) * FEATDIM);
        __builtin_prefetch(np + c4, 0, 3);
      }

      const float* xp = (col < nA) ? (srcA + (long)col * FEATDIM)
                                   : (srcB + (long)(col - nA) * FEATDIM);
      float4 x = *(const float4*)(xp + c4);

      float* dp = S + (long)row * FEATDIM + c4;
      atomAddF(dp + 0, v * x.x);
      atomAddF(dp + 1, v * x.y);
      atomAddF(dp + 2, v * x.z);
      atomAddF(dp + 3, v * x.w);
    }
    __syncthreads();   // protect LDS before next tile's staging
  }
}

// -------------------------------------------------------------------------
// Commit: dst += 2*max(S,0); rows < split -> outMain, else -> acc.
// Also re-zeroes S in the same pass (saves a separate 35.8MB zero pass
// before the next scatter; all buffers are L2-resident).
__global__ __launch_bounds__(256)
void commit_k(float* __restrict__ S, int nRows, int split,
              float* __restrict__ outMain, float* __restrict__ acc)
{
  long total = (long)nRows * 32;   // float4 chunks (128 floats / row)
  for (long i = (long)blockIdx.x * blockDim.x + threadIdx.x; i < total;
       i += (long)gridDim.x * blockDim.x) {
    int row = (int)(i >> 5);
    int c   = ((int)i & 31) * 4;
    float4* sp = (float4*)(S + (long)row * FEATDIM + c);
    float4 s = *sp;
    *sp = make_float4(0.f, 0.f, 0.f, 0.f);   // re-arm S for the next pass
    float4 h;
    h.x = 2.0f * fmaxf(s.x, 0.0f);
    h.y = 2.0f * fmaxf(s.y, 0.0f);
    h.z = 2.0f * fmaxf(s.z, 0.0f);
    h.w = 2.0f * fmaxf(s.w, 0.0f);
    float* d = (row < split) ? (outMain + (long)row * FEATDIM + c)
                             : (acc + (long)(row - split) * FEATDIM + c);
    float4 o = *(float4*)d;
    o.x += h.x; o.y += h.y; o.z += h.z; o.w += h.w;
    *(float4*)d = o;
  }
}

// -------------------------------------------------------------------------
// Zero a float buffer (n is a multiple of 4)
__global__ __launch_bounds__(256)
void zero_k(float* __restrict__ p, long n) {
  long n4 = n >> 2;
  float4* p4 = (float4*)p;
  for (long i = (long)blockIdx.x * blockDim.x + threadIdx.x; i < n4;
       i += (long)gridDim.x * blockDim.x) {
    p4[i] = make_float4(0.f, 0.f, 0.f, 0.f);
  }
}

// -------------------------------------------------------------------------
// out_m = inter*M1 + (1-inter)*M2
__global__ __launch_bounds__(256)
void combine_k(const float* __restrict__ M1, const float* __restrict__ M2,
               const float* __restrict__ interp, float* __restrict__ outm,
               long n)
{
  float t = interp[0];
  float u = 1.0f - t;
  long n4 = n >> 2;
  const float4* a4 = (const float4*)M1;
  const float4* b4 = (const float4*)M2;
  float4* o4 = (float4*)outm;
  for (long i = (long)blockIdx.x * blockDim.x + threadIdx.x; i < n4;
       i += (long)gridDim.x * blockDim.x) {
    float4 a = a4[i], b = b4[i], o;
    o.x = t * a.x + u * b.x;
    o.y = t * a.y + u * b.y;
    o.z = t * a.z + u * b.z;
    o.w = t * a.w + u * b.w;
    o4[i] = o;
  }
}

// -------------------------------------------------------------------------
static inline int gridFor(long work, int perBlock) {
  long g = (work + perBlock - 1) / perBlock;
  if (g < 1) g = 1;
  if (g > 32768) g = 32768;
  return (int)g;
}

extern "C" void kernel_launch(void* const* d_in, const int* in_sizes, int n_in,
                              void* d_out, int out_size, void* d_ws, size_t ws_size,
                              hipStream_t stream) {
  (void)in_sizes; (void)n_in; (void)out_size; (void)ws_size;

  const int*   a1r = (const int*)  d_in[0];
  const int*   a1c = (const int*)  d_in[1];
  const float* a1v = (const float*)d_in[2];
  const int*   a2r = (const int*)  d_in[3];
  const int*   a2c = (const int*)  d_in[4];
  const float* a2v = (const float*)d_in[5];
  const float* dE  = (const float*)d_in[6];
  const float* mE  = (const float*)d_in[7];
  const float* pE  = (const float*)d_in[8];
  const float* inter = (const float*)d_in[9];

  float* out   = (float*)d_out;
  float* out_m = out;                                 // 20000*128
  float* out_d = out + (long)MEDNUM  * FEATDIM;       // 50000*128
  float* out_p = out + (long)(MEDNUM + DIAGNUM) * FEATDIM; // 40000*128

  // workspace: S (N1*128 reused for both graphs) | M1 | M2
  float* S  = (float*)d_ws;
  float* M1 = S  + (long)N1 * FEATDIM;
  float* M2 = M1 + (long)MEDNUM * FEATDIM;

  const long nOutDP = (long)(DIAGNUM + PRONUM) * FEATDIM;  // out_d..out_p contiguous
  const long nM     = (long)MEDNUM * FEATDIM;

  // zero d/p output regions, both m-accumulators, and S (once; commit_k
  // re-zeroes S on every pass thereafter)
  zero_k<<<gridFor(nOutDP / 4, 256), 256, 0, stream>>>(out_d, nOutDP);
  zero_k<<<gridFor((2 * nM) / 4, 256), 256, 0, stream>>>(M1, 2 * nM);
  zero_k<<<gridFor(((long)N1 * FEATDIM) / 4, 256), 256, 0, stream>>>(
      S, (long)N1 * FEATDIM);

  // graph 1: two edge sets over e1 = [dEmbed ; mEmbed]
  for (int i = 0; i < 2; ++i) {
    spmm_scatter<<<gridFor(NNZ1, TILE), 256, 0, stream>>>(
        a1r + (long)i * NNZ1, a1c + (long)i * NNZ1, a1v + (long)i * NNZ1,
        NNZ1, dE, DIAGNUM, mE, S);
    commit_k<<<gridFor((long)N1 * 32, 256), 256, 0, stream>>>(
        S, N1, DIAGNUM, out_d, M1);   // also re-zeroes all N1 rows of S
  }

  // graph 2: two edge sets over e2 = [pEmbed ; mEmbed]
  for (int i = 0; i < 2; ++i) {
    spmm_scatter<<<gridFor(NNZ2, TILE), 256, 0, stream>>>(
        a2r + (long)i * NNZ2, a2c + (long)i * NNZ2, a2v + (long)i * NNZ2,
        NNZ2, pE, PRONUM, mE, S);
    commit_k<<<gridFor((long)N2 * 32, 256), 256, 0, stream>>>(
        S, N2, PRONUM, out_p, M2);    // re-zeroes first N2 rows of S
  }

  // out_m = inter*M1 + (1-inter)*M2
  combine_k<<<gridFor(nM / 4, 256), 256, 0, stream>>>(M1, M2, inter, out_m, nM);
}